// ProtoGtnet_78924319031945
// MI455X (gfx1250) — compile-verified
//
#include <hip/hip_runtime.h>
#include <hip/hip_bf16.h>
#include <cstddef>

// ---------------------------------------------------------------------------
// GTNet forward for MI455X (gfx1250).  Dominant compute: mixprop graph GEMM
// (1024x1024 adjacency x [1024, B*C*L] activations) -> V_WMMA_F32_16X16X4_F32,
// tiles fed via async global->LDS DMA (ASYNCcnt) when the toolchain has it.
// ---------------------------------------------------------------------------

#define NNODE   1024
#define BATCH   64
#define CRES    32
#define CSKIPCH 64
#define NODE_DIM 40
#define K_TOP   20
#define SEQ_T   12
#define RF_T    19
#define PARTS   64

typedef __attribute__((ext_vector_type(2))) float v2f;
typedef __attribute__((ext_vector_type(8))) float v8f;
typedef __attribute__((ext_vector_type(4))) int v4i;

#define LDS_AS __attribute__((address_space(3)))
#define GLB_AS __attribute__((address_space(1)))

#if __has_builtin(__builtin_amdgcn_global_load_async_to_lds_b32)
#define HAVE_ASYNC_B32 1
#else
#define HAVE_ASYNC_B32 0
#endif
#if __has_builtin(__builtin_amdgcn_global_load_async_to_lds_b128)
#define HAVE_ASYNC_B128 1
#else
#define HAVE_ASYNC_B128 0
#endif

static __device__ __forceinline__ void wait_async0() {
#if __has_builtin(__builtin_amdgcn_s_wait_asynccnt)
  __builtin_amdgcn_s_wait_asynccnt(0);
#else
  asm volatile("s_wait_asynccnt 0" ::: "memory");
#endif
}

static __device__ __forceinline__ float sigmoidf_(float x) {
  return 1.0f / (1.0f + expf(-x));
}

// ------------------------------ adjacency ----------------------------------

__global__ void k_nodevec(const float* __restrict__ emb, const int* __restrict__ idx,
                          const float* __restrict__ W, const float* __restrict__ bias,
                          float* __restrict__ nv) {
  int tid = blockIdx.x * blockDim.x + threadIdx.x;
  if (tid >= NNODE * NODE_DIM) return;
  int v = tid / NODE_DIM, d = tid % NODE_DIM;
  const float* er = emb + (size_t)idx[v] * NODE_DIM;
  const float* wr = W + (size_t)d * NODE_DIM;
  float s = bias[d];
  for (int e = 0; e < NODE_DIM; ++e) s += er[e] * wr[e];
  nv[tid] = tanhf(3.0f * s);
}

__global__ void k_scores(const float* __restrict__ nv1, const float* __restrict__ nv2,
                         const float* __restrict__ noise,
                         float* __restrict__ adj, float* __restrict__ score) {
  int tid = blockIdx.x * blockDim.x + threadIdx.x;
  if (tid >= NNODE * NNODE) return;
  int v = tid / NNODE, w = tid % NNODE;
  const float* p1 = nv1 + (size_t)v * NODE_DIM;
  const float* p2 = nv2 + (size_t)w * NODE_DIM;
  const float* q1 = nv2 + (size_t)v * NODE_DIM;
  const float* q2 = nv1 + (size_t)w * NODE_DIM;
  float s = 0.f;
  for (int d = 0; d < NODE_DIM; ++d) s += p1[d] * p2[d] - q1[d] * q2[d];
  float t = tanhf(3.0f * s);
  float a = t > 0.f ? t : 0.f;
  adj[tid] = a;
  score[tid] = a + 0.01f * noise[tid];
}

// one thread per row: selection of top-K (first-max tie break, like lax.top_k)
__global__ void k_topk(const float* __restrict__ adj, const float* __restrict__ score,
                       float* __restrict__ am) {
  int v = blockIdx.x * blockDim.x + threadIdx.x;
  if (v >= NNODE) return;
  const float* srow = score + (size_t)v * NNODE;
  const float* arow = adj + (size_t)v * NNODE;
  float* orow = am + (size_t)v * NNODE;
  for (int w = 0; w < NNODE; ++w) orow[w] = 0.f;
  int chosen[K_TOP];
  for (int j = 0; j < K_TOP; ++j) {
    float best = -3.0e38f; int bi = 0;
    for (int w = 0; w < NNODE; ++w) {
      bool used = false;
      for (int j2 = 0; j2 < j; ++j2) used = used || (chosen[j2] == w);
      if (used) continue;
      float s = srow[w];
      if (s > best) { best = s; bi = w; }
    }
    chosen[j] = bi;
    orow[bi] = arow[bi];
  }
}

__global__ void k_sums(const float* __restrict__ am,
                       float* __restrict__ rsum, float* __restrict__ csum) {
  int v = blockIdx.x * blockDim.x + threadIdx.x;
  if (v >= NNODE) return;
  float r = 1.f, c = 1.f;  // +1 for identity
  for (int w = 0; w < NNODE; ++w) {
    r += am[(size_t)v * NNODE + w];
    c += am[(size_t)w * NNODE + v];
  }
  rsum[v] = r; csum[v] = c;
}

__global__ void k_normadj(const float* __restrict__ am,
                          const float* __restrict__ rsum, const float* __restrict__ csum,
                          float* __restrict__ a1, float* __restrict__ a2) {
  int tid = blockIdx.x * blockDim.x + threadIdx.x;
  if (tid >= NNODE * NNODE) return;
  int v = tid / NNODE, w = tid % NNODE;
  float e = (v == w) ? 1.f : 0.f;
  a1[tid] = (am[(size_t)v * NNODE + w] + e) / rsum[v];
  a2[tid] = (am[(size_t)w * NNODE + v] + e) / csum[v];
}

// ------------------------------ front of net -------------------------------

__global__ void k_start(const float* __restrict__ x, const float* __restrict__ w,
                        const float* __restrict__ b, float* __restrict__ xr) {
  long tid = (long)blockIdx.x * blockDim.x + threadIdx.x;
  long total = (long)BATCH * NNODE * RF_T;
  if (tid >= total) return;
  int t = tid % RF_T; long r = tid / RF_T;
  int n = r % NNODE; int bb = (int)(r / NNODE);
  float in0 = 0.f, in1 = 0.f;
  if (t >= RF_T - SEQ_T) {
    int tt = t - (RF_T - SEQ_T);
    in0 = x[(((size_t)bb * 2 + 0) * NNODE + n) * SEQ_T + tt];
    in1 = x[(((size_t)bb * 2 + 1) * NNODE + n) * SEQ_T + tt];
  }
  for (int c = 0; c < CRES; ++c) {
    xr[(((size_t)bb * CRES + c) * NNODE + n) * RF_T + t] =
        b[c] + w[c * 2 + 0] * in0 + w[c * 2 + 1] * in1;
  }
}

__global__ void k_skip0(const float* __restrict__ x, const float* __restrict__ w,
                        const float* __restrict__ b, float* __restrict__ skip) {
  long tid = (long)blockIdx.x * blockDim.x + threadIdx.x;
  long total = (long)BATCH * CSKIPCH * NNODE;
  if (tid >= total) return;
  int n = tid % NNODE; long r = tid / NNODE;
  int c = r % CSKIPCH; int bb = (int)(r / CSKIPCH);
  float s = b[c];
  for (int ci = 0; ci < 2; ++ci)
    for (int t = 0; t < SEQ_T; ++t)
      s += w[(c * 2 + ci) * RF_T + (t + RF_T - SEQ_T)] *
           x[(((size_t)bb * 2 + ci) * NNODE + n) * SEQ_T + t];
  skip[((size_t)bb * CSKIPCH + c) * NNODE + n] = s;
}

// --------------------------- inception + gate ------------------------------

struct IncArgs {
  const float* fw[4]; const float* fb[4];
  const float* gw[4]; const float* gb[4];
};

__global__ void k_inception(const float* __restrict__ xr, IncArgs ia,
                            int Tin, int Tout, float* __restrict__ xc) {
  long tid = (long)blockIdx.x * blockDim.x + threadIdx.x;
  long total = (long)BATCH * CRES * NNODE * Tout;
  if (tid >= total) return;
  int t = tid % Tout; long r = tid / Tout;
  int n = r % NNODE; r /= NNODE;
  int cc = r % CRES; int bb = (int)(r / CRES);
  int br = cc >> 3, c8 = cc & 7;
  int k = (br == 0) ? 2 : (br == 1) ? 3 : (br == 2) ? 6 : 7;
  int off = t + 7 - k;  // window always ends at t+6
  const float* fw = ia.fw[br]; const float* gw = ia.gw[br];
  float f = ia.fb[br][c8], g = ia.gb[br][c8];
  for (int ci = 0; ci < CRES; ++ci) {
    const float* xp = xr + (((size_t)bb * CRES + ci) * NNODE + n) * Tin + off;
    const float* fwp = fw + (size_t)(c8 * CRES + ci) * k;
    const float* gwp = gw + (size_t)(c8 * CRES + ci) * k;
    for (int j = 0; j < k; ++j) { float xv = xp[j]; f += fwp[j] * xv; g += gwp[j] * xv; }
  }
  xc[tid] = tanhf(f) * sigmoidf_(g);
}

__global__ void k_skipc(const float* __restrict__ xc, const float* __restrict__ w,
                        const float* __restrict__ b, float* __restrict__ skip, int Tout) {
  long tid = (long)blockIdx.x * blockDim.x + threadIdx.x;
  long total = (long)BATCH * CSKIPCH * NNODE;
  if (tid >= total) return;
  int n = tid % NNODE; long r = tid / NNODE;
  int c = r % CSKIPCH; int bb = (int)(r / CSKIPCH);
  float s = b[c];
  for (int ci = 0; ci < CRES; ++ci)
    for (int t = 0; t < Tout; ++t)
      s += w[(c * CRES + ci) * Tout + t] *
           xc[(((size_t)bb * CRES + ci) * NNODE + n) * Tout + t];
  skip[((size_t)bb * CSKIPCH + c) * NNODE + n] += s;
}

// ------------------------------ mixprop ------------------------------------

__global__ void k_init_xg(const float* __restrict__ b1, const float* __restrict__ b2,
                          float* __restrict__ xg, int Lt) {
  long tid = (long)blockIdx.x * blockDim.x + threadIdx.x;
  long total = (long)BATCH * CRES * NNODE * Lt;
  if (tid >= total) return;
  int c = (int)((tid / ((long)NNODE * Lt)) % CRES);
  xg[tid] = b1[c] + b2[c];
}

// Hout[v,m] = 0.05*Xin[v,m] + 0.95 * sum_w A[v,w]*Hin[w,m]
// m enumerates (bc = b*32+c, l);  H layout [b][c][node][l].
// Block: 256 thr (8 waves, 4Mx2N) -> 64x32 tile, K staged in LDS chunks of 32.
// A stride 36: 16B-aligned float4 stores AND conflict-free fragment reads
// (gcd(36,64)=4 -> 16 distinct banks over the 16-lane column read).
__global__ __launch_bounds__(256)
void k_prop(const float* __restrict__ A, const float* __restrict__ Hin,
            const float* __restrict__ Xin, float* __restrict__ Hout,
            int Lt, int Mtot) {
  __shared__ float As[64][36];
  __shared__ float Bs[32][33];
  __shared__ long long colOff[32];   // element offset of column base, -1 if OOB
  const int tid = threadIdx.x;
  const int v0 = blockIdx.y * 64;
  const int m0 = blockIdx.x * 32;
  const int wid = tid >> 5, lane = tid & 31;
  const int wv = wid & 3, wm = wid >> 2;
  const int lrow = lane & 15, lhi = lane >> 4;

  if (tid < 32) {
    int m = m0 + tid;
    long long off = -1;
    if (m < Mtot) {
      int bc = m / Lt, l = m - bc * Lt;
      off = (long long)bc * NNODE * Lt + l;
    }
    colOff[tid] = off;
  }
  __syncthreads();

  const int ar0 = tid >> 3;          // A: row 0..31 (+32 on 2nd quad)
  const int ac4 = (tid & 7) * 4;     // A: col {0,4,...,28}

  v8f acc = {};
  for (int k0 = 0; k0 < NNODE; k0 += 32) {
    // ---- stage A tile 64x32 ----
#if HAVE_ASYNC_B128
#pragma unroll
    for (int j = 0; j < 2; ++j) {
      int r = ar0 + j * 32;
      float* gp = const_cast<float*>(&A[(size_t)(v0 + r) * NNODE + (k0 + ac4)]);
      __builtin_amdgcn_global_load_async_to_lds_b128(
          (GLB_AS v4i*)gp, (LDS_AS v4i*)&As[r][ac4], 0, 0);
    }
#else
#pragma unroll
    for (int j = 0; j < 2; ++j) {
      int r = ar0 + j * 32;
      const float4 av = *(const float4*)&A[(size_t)(v0 + r) * NNODE + (k0 + ac4)];
      *(float4*)&As[r][ac4] = av;
    }
#endif
    // ---- stage B tile 32x32 ----
#if HAVE_ASYNC_B32
#pragma unroll
    for (int j = 0; j < 4; ++j) {
      int i = tid + j * 256;
      int r = i >> 5, c = i & 31;
      long long off = colOff[c];
      if (off >= 0) {
        float* gp = const_cast<float*>(Hin + off + (size_t)(k0 + r) * Lt);
        __builtin_amdgcn_global_load_async_to_lds_b32(
            (GLB_AS int*)gp, (LDS_AS int*)&Bs[r][c], 0, 0);
      }
    }
    wait_async0();
#else
    {
      float bv[4];
#pragma unroll
      for (int j = 0; j < 4; ++j) {
        int i = tid + j * 256;
        int r = i >> 5, c = i & 31;
        long long off = colOff[c];
        bv[j] = (off >= 0) ? Hin[off + (size_t)(k0 + r) * Lt] : 0.f;
      }
#pragma unroll
      for (int j = 0; j < 4; ++j) {
        int i = tid + j * 256;
        Bs[i >> 5][i & 31] = bv[j];
      }
    }
#endif
#if HAVE_ASYNC_B128 && !HAVE_ASYNC_B32
    wait_async0();
#endif
    __syncthreads();
#pragma unroll
    for (int kk = 0; kk < 32; kk += 4) {
      v2f a, b;
      // A 16x4 fp32 fragment: lanes 0-15 -> K={kk,kk+1}, lanes 16-31 -> K={kk+2,kk+3}
      a[0] = As[wv * 16 + lrow][kk + lhi * 2 + 0];
      a[1] = As[wv * 16 + lrow][kk + lhi * 2 + 1];
      b[0] = Bs[kk + lhi * 2 + 0][wm * 16 + lrow];
      b[1] = Bs[kk + lhi * 2 + 1][wm * 16 + lrow];
      acc = __builtin_amdgcn_wmma_f32_16x16x4_f32(false, a, false, b, (short)0, acc,
                                                  false, false);
    }
    __syncthreads();
  }
  // C/D 16x16 layout: vgpr g -> M = g (+8 for lanes 16-31), N = lane&15
  long long coff = colOff[wm * 16 + lrow];
  if (coff >= 0) {
#pragma unroll
    for (int g = 0; g < 8; ++g) {
      int rr = v0 + wv * 16 + g + lhi * 8;
      size_t addr = (size_t)coff + (size_t)rr * Lt;
      Hout[addr] = 0.05f * Xin[addr] + 0.95f * acc[g];
    }
  }
}

// xg[b,co,n,l] += W_chunk[co,:] . H[b,:,n,l]   (chunk of the 96-in 1x1 conv)
__global__ __launch_bounds__(256)
void k_mixconv_acc(const float* __restrict__ W, int chunk,
                   const float* __restrict__ H, float* __restrict__ xg, int Lt) {
  __shared__ float Ws[CRES * CRES];
  for (int i = threadIdx.x; i < CRES * CRES; i += blockDim.x)
    Ws[i] = W[(i >> 5) * (3 * CRES) + chunk * CRES + (i & 31)];
  __syncthreads();
  long tid = (long)blockIdx.x * blockDim.x + threadIdx.x;
  long total = (long)BATCH * NNODE * Lt;
  if (tid >= total) return;
  int l = tid % Lt; long r = tid / Lt;
  int n = r % NNODE; int bb = (int)(r / NNODE);
  float hin[CRES];
  for (int ci = 0; ci < CRES; ++ci)
    hin[ci] = H[(((size_t)bb * CRES + ci) * NNODE + n) * Lt + l];
  for (int co = 0; co < CRES; ++co) {
    float s = 0.f;
    for (int ci = 0; ci < CRES; ++ci) s += Ws[co * CRES + ci] * hin[ci];
    size_t o = (((size_t)bb * CRES + co) * NNODE + n) * Lt + l;
    xg[o] += s;
  }
}

// ------------------------- residual + layernorm ----------------------------

__global__ void k_resadd(float* __restrict__ xg, const float* __restrict__ xr,
                         int Tin, int Lt) {
  long tid = (long)blockIdx.x * blockDim.x + threadIdx.x;
  long total = (long)BATCH * CRES * NNODE * Lt;
  if (tid >= total) return;
  int l = tid % Lt; long r = tid / Lt;
  int n = r % NNODE; r /= NNODE;
  int c = r % CRES; int bb = (int)(r / CRES);
  xg[tid] += xr[(((size_t)bb * CRES + c) * NNODE + n) * Tin + (Tin - Lt + l)];
}

__global__ __launch_bounds__(256)
void k_stats_part(const float* __restrict__ xg, float* __restrict__ part,
                  int Lt, int parts) {
  __shared__ float s1[256], s2[256];
  int b = blockIdx.y, p = blockIdx.x;
  size_t cnt = (size_t)CRES * NNODE * Lt;
  const float* base = xg + (size_t)b * cnt;
  float a = 0.f, q = 0.f;
  for (size_t i = (size_t)p * 256 + threadIdx.x; i < cnt; i += (size_t)parts * 256) {
    float v = base[i]; a += v; q += v * v;
  }
  s1[threadIdx.x] = a; s2[threadIdx.x] = q;
  __syncthreads();
  for (int s = 128; s > 0; s >>= 1) {
    if ((int)threadIdx.x < s) {
      s1[threadIdx.x] += s1[threadIdx.x + s];
      s2[threadIdx.x] += s2[threadIdx.x + s];
    }
    __syncthreads();
  }
  if (threadIdx.x == 0) {
    part[(b * parts + p) * 2 + 0] = s1[0];
    part[(b * parts + p) * 2 + 1] = s2[0];
  }
}

__global__ void k_stats_final(const float* __restrict__ part, float* __restrict__ stat,
                              int Lt, int parts) {
  int b = blockIdx.x;
  if (threadIdx.x != 0) return;
  double s = 0.0, q = 0.0;
  for (int p = 0; p < parts; ++p) {
    s += (double)part[(b * parts + p) * 2 + 0];
    q += (double)part[(b * parts + p) * 2 + 1];
  }
  double cnt = (double)CRES * NNODE * Lt;
  double m = s / cnt;
  double var = q / cnt - m * m;
  stat[b * 2 + 0] = (float)m;
  stat[b * 2 + 1] = (float)(1.0 / sqrt(var + 1e-5));
}

__global__ void k_lnapply(const float* __restrict__ xg, const float* __restrict__ stat,
                          const float* __restrict__ nw, const float* __restrict__ nb,
                          float* __restrict__ xrout, int Lt) {
  long tid = (long)blockIdx.x * blockDim.x + threadIdx.x;
  long total = (long)BATCH * CRES * NNODE * Lt;
  if (tid >= total) return;
  long per = (long)CRES * NNODE * Lt;
  int bb = (int)(tid / per);
  long rem = tid - (long)bb * per;
  float m = stat[bb * 2 + 0], r = stat[bb * 2 + 1];
  xrout[tid] = (xg[tid] - m) * r * nw[rem] + nb[rem];
}

// ------------------------------- head --------------------------------------

__global__ __launch_bounds__(128)
void k_final(const float* __restrict__ xr, const float* __restrict__ skip,
             const float* __restrict__ sew, const float* __restrict__ seb,
             const float* __restrict__ e1w, const float* __restrict__ e1b,
             const float* __restrict__ e2w, const float* __restrict__ e2b,
             float* __restrict__ out) {
  __shared__ float s[CSKIPCH];
  __shared__ float e1[128];
  int bn = blockIdx.x;
  int n = bn % NNODE, bb = bn / NNODE;
  int t = threadIdx.x;
  if (t < CSKIPCH) {
    float v = seb[t] + skip[((size_t)bb * CSKIPCH + t) * NNODE + n];
    for (int ci = 0; ci < CRES; ++ci)
      v += sew[t * CRES + ci] * xr[((size_t)bb * CRES + ci) * NNODE + n];  // T==1
    s[t] = v > 0.f ? v : 0.f;
  }
  __syncthreads();
  {
    float v = e1b[t];
    for (int k = 0; k < CSKIPCH; ++k) v += e1w[t * CSKIPCH + k] * s[k];
    e1[t] = v > 0.f ? v : 0.f;
  }
  __syncthreads();
  if (t < 12) {
    float o = e2b[t];
    for (int k = 0; k < 128; ++k) o += e2w[t * 128 + k] * e1[k];
    out[((size_t)bb * 12 + t) * NNODE + n] = o;
  }
}

// ----------------------------- host side -----------------------------------

// Input leaf order: jax pytree flatten (dict keys sorted, lists in order).
enum {
  IN_IDX = 0, IN_NOISE = 1, IN_EMB1 = 2, IN_EMB2 = 3,
  IN_END1_B = 4, IN_END1_W = 5, IN_END2_B = 6, IN_END2_W = 7,
  IN_FILTER = 8,   // [layer][branch]{b,w}: 24 leaves
  IN_G1 = 32,      // [layer]{b,w}: 6
  IN_G2 = 38,
  IN_GATE = 44,    // 24
  IN_LIN1_B = 68, IN_LIN1_W = 69, IN_LIN2_B = 70, IN_LIN2_W = 71,
  IN_NORM = 72,    // [layer]{b,w}: 6
  IN_SKIP0_B = 78, IN_SKIP0_W = 79,
  IN_SKIPE_B = 80, IN_SKIPE_W = 81,
  IN_SKIPC = 82,   // 6
  IN_START_B = 88, IN_START_W = 89,
  IN_X = 90
};

static inline unsigned cdiv(long a, int b) { return (unsigned)((a + b - 1) / b); }

extern "C" void kernel_launch(void* const* d_in, const int* in_sizes, int n_in,
                              void* d_out, int out_size, void* d_ws, size_t ws_size,
                              hipStream_t stream) {
  (void)in_sizes; (void)n_in; (void)out_size; (void)ws_size;
  const int*   idx   = (const int*)d_in[IN_IDX];
  const float* noise = (const float*)d_in[IN_NOISE];
  const float* x     = (const float*)d_in[IN_X];

  // workspace carve (floats)
  float* ws = (float*)d_ws;
  size_t off = 0;
  auto alloc = [&](size_t n) { float* p = ws + off; off += n; return p; };
  float* nv1  = alloc((size_t)NNODE * NODE_DIM);
  float* nv2  = alloc((size_t)NNODE * NODE_DIM);
  float* adj  = alloc((size_t)NNODE * NNODE);
  float* scr  = alloc((size_t)NNODE * NNODE);
  float* am   = alloc((size_t)NNODE * NNODE);
  float* a1   = alloc((size_t)NNODE * NNODE);
  float* a2   = alloc((size_t)NNODE * NNODE);
  float* rsum = alloc(NNODE);
  float* csum = alloc(NNODE);
  // keep 16B alignment for the GEMM buffers
  off = (off + 3) & ~(size_t)3;
  float* xr   = alloc((size_t)BATCH * CRES * NNODE * RF_T);
  off = (off + 3) & ~(size_t)3;
  float* xc   = alloc((size_t)BATCH * CRES * NNODE * 13);
  float* h1   = alloc((size_t)BATCH * CRES * NNODE * 13);
  float* h2   = alloc((size_t)BATCH * CRES * NNODE * 13);
  float* xg   = alloc((size_t)BATCH * CRES * NNODE * 13);
  float* skip = alloc((size_t)BATCH * CSKIPCH * NNODE);
  float* part = alloc((size_t)BATCH * PARTS * 2);
  float* stat = alloc((size_t)BATCH * 2);

  // ---- adjacency ----
  k_nodevec<<<cdiv(NNODE * NODE_DIM, 256), 256, 0, stream>>>(
      (const float*)d_in[IN_EMB1], idx, (const float*)d_in[IN_LIN1_W],
      (const float*)d_in[IN_LIN1_B], nv1);
  k_nodevec<<<cdiv(NNODE * NODE_DIM, 256), 256, 0, stream>>>(
      (const float*)d_in[IN_EMB2], idx, (const float*)d_in[IN_LIN2_W],
      (const float*)d_in[IN_LIN2_B], nv2);
  k_scores<<<cdiv((long)NNODE * NNODE, 256), 256, 0, stream>>>(nv1, nv2, noise, adj, scr);
  k_topk<<<cdiv(NNODE, 256), 256, 0, stream>>>(adj, scr, am);
  k_sums<<<cdiv(NNODE, 256), 256, 0, stream>>>(am, rsum, csum);
  k_normadj<<<cdiv((long)NNODE * NNODE, 256), 256, 0, stream>>>(am, rsum, csum, a1, a2);

  // ---- front ----
  k_start<<<cdiv((long)BATCH * NNODE * RF_T, 256), 256, 0, stream>>>(
      x, (const float*)d_in[IN_START_W], (const float*)d_in[IN_START_B], xr);
  k_skip0<<<cdiv((long)BATCH * CSKIPCH * NNODE, 256), 256, 0, stream>>>(
      x, (const float*)d_in[IN_SKIP0_W], (const float*)d_in[IN_SKIP0_B], skip);

  const int TinA[3] = {19, 13, 7};
  const int TouA[3] = {13, 7, 1};

  for (int i = 0; i < 3; ++i) {
    const int Ti = TinA[i], To = TouA[i];
    IncArgs ia;
    for (int br = 0; br < 4; ++br) {
      ia.fb[br] = (const float*)d_in[IN_FILTER + (i * 4 + br) * 2 + 0];
      ia.fw[br] = (const float*)d_in[IN_FILTER + (i * 4 + br) * 2 + 1];
      ia.gb[br] = (const float*)d_in[IN_GATE + (i * 4 + br) * 2 + 0];
      ia.gw[br] = (const float*)d_in[IN_GATE + (i * 4 + br) * 2 + 1];
    }
    const float* g1b = (const float*)d_in[IN_G1 + i * 2 + 0];
    const float* g1w = (const float*)d_in[IN_G1 + i * 2 + 1];
    const float* g2b = (const float*)d_in[IN_G2 + i * 2 + 0];
    const float* g2w = (const float*)d_in[IN_G2 + i * 2 + 1];
    const float* scb = (const float*)d_in[IN_SKIPC + i * 2 + 0];
    const float* scw = (const float*)d_in[IN_SKIPC + i * 2 + 1];
    const float* nb  = (const float*)d_in[IN_NORM + i * 2 + 0];
    const float* nw  = (const float*)d_in[IN_NORM + i * 2 + 1];

    long elems = (long)BATCH * CRES * NNODE * To;
    k_inception<<<cdiv(elems, 256), 256, 0, stream>>>(xr, ia, Ti, To, xc);
    k_skipc<<<cdiv((long)BATCH * CSKIPCH * NNODE, 256), 256, 0, stream>>>(
        xc, scw, scb, skip, To);

    k_init_xg<<<cdiv(elems, 256), 256, 0, stream>>>(g1b, g2b, xg, To);
    const int Mtot = BATCH * CRES * To;  // columns (bc,l)
    dim3 pg(cdiv(Mtot, 32), NNODE / 64);
    long bnl = (long)BATCH * NNODE * To;
    for (int d = 0; d < 2; ++d) {
      const float* Ad = d ? a2 : a1;
      const float* Wd = d ? g2w : g1w;
      k_mixconv_acc<<<cdiv(bnl, 256), 256, 0, stream>>>(Wd, 0, xc, xg, To);
      k_prop<<<pg, 256, 0, stream>>>(Ad, xc, xc, h1, To, Mtot);
      k_mixconv_acc<<<cdiv(bnl, 256), 256, 0, stream>>>(Wd, 1, h1, xg, To);
      k_prop<<<pg, 256, 0, stream>>>(Ad, h1, xc, h2, To, Mtot);
      k_mixconv_acc<<<cdiv(bnl, 256), 256, 0, stream>>>(Wd, 2, h2, xg, To);
    }

    k_resadd<<<cdiv(elems, 256), 256, 0, stream>>>(xg, xr, Ti, To);
    k_stats_part<<<dim3(PARTS, BATCH), 256, 0, stream>>>(xg, part, To, PARTS);
    k_stats_final<<<BATCH, 1, 0, stream>>>(part, stat, To, PARTS);
    k_lnapply<<<cdiv(elems, 256), 256, 0, stream>>>(xg, stat, nw, nb, xr, To);
  }

  k_final<<<BATCH * NNODE, 128, 0, stream>>>(
      xr, skip, (const float*)d_in[IN_SKIPE_W], (const float*)d_in[IN_SKIPE_B],
      (const float*)d_in[IN_END1_W], (const float*)d_in[IN_END1_B],
      (const float*)d_in[IN_END2_W], (const float*)d_in[IN_END2_B], (float*)d_out);
}